// GeometricConsistencyPretrain_86431921865227
// MI455X (gfx1250) — compile-verified
//
#include <hip/hip_runtime.h>
#include <math.h>

#define BB   2
#define NN   8192
#define BN   (BB * NN)
#define KNN  8
#define EE   384
#define H1   256
#define H2   128
#define EPSV 1e-8f

// LDS row strides (floats): even (8B-aligned float2 loads) and stride%64==2 -> conflict free
#define FS   386
#define H1S  258
#define H2S  130
#define W2S  130   // w_d2 LDS stride: 128 row + 2 pad dwords (TDM pad feature)

typedef float v8f __attribute__((ext_vector_type(8)));
typedef float v2f __attribute__((ext_vector_type(2)));
typedef unsigned int u32;
typedef u32 u32x4 __attribute__((ext_vector_type(4)));
typedef u32 u32x8 __attribute__((ext_vector_type(8)));

// ---------------------------------------------------------------------------
// Kernel 0: zero the 4 loss accumulators (must happen every launch: graph replay)
// ---------------------------------------------------------------------------
__global__ void gcp_init_kernel(float* __restrict__ accum) {
    if (threadIdx.x < 4) accum[threadIdx.x] = 0.0f;
}

// ---------------------------------------------------------------------------
// Kernel 1: KNN (top-8 by distance, includes self) + curvature-std mask weight
// One thread per query point; candidates staged through LDS tiles of 128.
// 128-thread blocks (4 waves) -> 128 workgroups spread across WGPs.
// d2 uses the reference formula sq_i + sq_j - 2*dot to match its fp ordering.
// ---------------------------------------------------------------------------
#define KT 128
__global__ __launch_bounds__(KT) void gcp_knn_kernel(
    const float* __restrict__ points, const float* __restrict__ mask_rand,
    int* __restrict__ knn_idx, float* __restrict__ wmask)
{
    __shared__ float sx[KT], sy[KT], sz[KT], sq[KT];

    const int t  = threadIdx.x;
    const int gi = blockIdx.x * KT + t;           // 0..BN-1
    const int b  = gi / NN;
    const float* pb = points + (size_t)b * NN * 3;

    const float px = points[gi * 3 + 0];
    const float py = points[gi * 3 + 1];
    const float pz = points[gi * 3 + 2];
    const float sqi = px * px + py * py + pz * pz;

    float bd[KNN];
    int   bi[KNN];
#pragma unroll
    for (int k = 0; k < KNN; ++k) { bd[k] = 3.4e38f; bi[k] = 0; }

    for (int tile = 0; tile < NN; tile += KT) {
        __syncthreads();
        const int j = tile + t;
        const float cx = pb[j * 3 + 0];
        const float cy = pb[j * 3 + 1];
        const float cz = pb[j * 3 + 2];
        sx[t] = cx; sy[t] = cy; sz[t] = cz;
        sq[t] = cx * cx + cy * cy + cz * cz;
        __syncthreads();

        for (int c = 0; c < KT; ++c) {
            const float d2 = sqi + sq[c] - 2.0f * (px * sx[c] + py * sy[c] + pz * sz[c]);
            if (d2 < bd[KNN - 1]) {
                bd[KNN - 1] = d2;
                bi[KNN - 1] = tile + c;
#pragma unroll
                for (int k = KNN - 1; k > 0; --k) {
                    if (bd[k] < bd[k - 1]) {      // strict <: keeps lower index first on ties
                        float td = bd[k]; bd[k] = bd[k - 1]; bd[k - 1] = td;
                        int   ti = bi[k]; bi[k] = bi[k - 1]; bi[k - 1] = ti;
                    }
                }
            }
        }
    }

    // curv_std over sqrt distances, ddof=1
    float d[KNN], mean = 0.0f;
#pragma unroll
    for (int k = 0; k < KNN; ++k) {
        d[k] = sqrtf(fmaxf(bd[k], 0.0f));
        mean += d[k];
    }
    mean *= (1.0f / KNN);
    float var = 0.0f;
#pragma unroll
    for (int k = 0; k < KNN; ++k) { float e = d[k] - mean; var += e * e; }
    var *= (1.0f / (KNN - 1));
    const float cstd = sqrtf(var);

    const float prob = 0.6f / (1.0f + expf(-5.0f * (cstd - 0.2f)));
    wmask[gi] = (mask_rand[gi] < prob) ? 1.0f : 0.0f;

#pragma unroll
    for (int k = 0; k < KNN; ++k) knn_idx[gi * KNN + k] = bi[k];
}

// ---------------------------------------------------------------------------
// Kernel 2: centroid of 8 neighbors -> normal = (p - c)/(|p-c| + eps)
// ---------------------------------------------------------------------------
__global__ __launch_bounds__(128) void gcp_normals_kernel(
    const float* __restrict__ points, const int* __restrict__ knn_idx,
    float* __restrict__ normals)
{
    const int gi = blockIdx.x * 128 + threadIdx.x;
    const int b  = gi / NN;
    const float* pb = points + (size_t)b * NN * 3;

    float cx = 0.0f, cy = 0.0f, cz = 0.0f;
#pragma unroll
    for (int k = 0; k < KNN; ++k) {
        const int j = knn_idx[gi * KNN + k];
        cx += pb[j * 3 + 0]; cy += pb[j * 3 + 1]; cz += pb[j * 3 + 2];
    }
    cx *= 0.125f; cy *= 0.125f; cz *= 0.125f;

    const float dx = points[gi * 3 + 0] - cx;
    const float dy = points[gi * 3 + 1] - cy;
    const float dz = points[gi * 3 + 2] - cz;
    const float inv = 1.0f / (sqrtf(dx * dx + dy * dy + dz * dz) + EPSV);
    normals[gi * 3 + 0] = dx * inv;
    normals[gi * 3 + 1] = dy * inv;
    normals[gi * 3 + 2] = dz * inv;
}

// ---------------------------------------------------------------------------
// Kernel 3: curvature = mean_k (1 - dot(n_k, n))
// ---------------------------------------------------------------------------
__global__ __launch_bounds__(128) void gcp_curv_kernel(
    const float* __restrict__ normals, const int* __restrict__ knn_idx,
    float* __restrict__ curv)
{
    const int gi = blockIdx.x * 128 + threadIdx.x;
    const int b  = gi / NN;
    const float* nb = normals + (size_t)b * NN * 3;

    const float nx = normals[gi * 3 + 0];
    const float ny = normals[gi * 3 + 1];
    const float nz = normals[gi * 3 + 2];

    float acc = 0.0f;
#pragma unroll
    for (int k = 0; k < KNN; ++k) {
        const int j = knn_idx[gi * KNN + k];
        acc += 1.0f - (nb[j * 3 + 0] * nx + nb[j * 3 + 1] * ny + nb[j * 3 + 2] * nz);
    }
    curv[gi] = acc * (1.0f / KNN);
}

// ---------------------------------------------------------------------------
// Kernel 4: fused MLP (3->384->256->128->4) + loss partials for a 16-row tile.
// Heavy layers use V_WMMA_F32_16X16X4_F32 (native fp32 matrix pipe).
// w_d2 (256x128 f32) is prefetched into LDS by the Tensor Data Mover at kernel
// entry (one TDM op, wave 0), overlapping the DMA with layers 1-2; layer 3
// then reads B-fragments from LDS. TDM pad feature inserts 2 dwords per
// 128-dword row -> stride 130 -> bank-conflict-free.
// Block = 256 threads = 8 waves; each wave owns 16x16 output tiles.
// ---------------------------------------------------------------------------
__global__ __launch_bounds__(256) void gcp_mlp_loss_kernel(
    const float* __restrict__ points,
    const float* __restrict__ w_enc, const float* __restrict__ b_enc,
    const float* __restrict__ w_d1,  const float* __restrict__ b_d1,
    const float* __restrict__ w_d2,  const float* __restrict__ b_d2,
    const float* __restrict__ w_d3,  const float* __restrict__ b_d3,
    const float* __restrict__ wmask, const float* __restrict__ normals,
    const float* __restrict__ curv,  float* __restrict__ accum)
{
    __shared__ __align__(16) float sW2[H1 * W2S];   // 256 x 130 = 130 KB (TDM dest)
    __shared__ __align__(16) float sP [16 * 3];
    __shared__ __align__(16) float sF [16 * FS];
    __shared__ __align__(16) float sH1[16 * H1S];
    __shared__ __align__(16) float sH2[16 * H2S];
    __shared__ float sRec[16 * 4];
    __shared__ float sRed[4 * 16];

    const int t    = threadIdx.x;
    const int row0 = blockIdx.x * 16;

    // ---- TDM: async prefetch of w_d2 into LDS (wave 0 issues one op) ----
    if (t < 32) {
        const unsigned long long ga = (unsigned long long)(uintptr_t)w_d2;
        const u32 lds_base = (u32)(uintptr_t)&sW2[0];
        u32x4 g0;
        g0[0] = 1u;                                              // count=1, user descriptor
        g0[1] = lds_base;                                        // lds_addr [63:32]
        g0[2] = (u32)ga;                                         // global_addr lo [95:64]
        g0[3] = ((u32)(ga >> 32) & 0x01FFFFFFu) | 0x80000000u;   // ga hi [120:96] | type=2
        u32x8 g1;
        g1[0] = (2u << 16)            // data_size = 4 bytes
              | (1u << 20)            // pad_enable
              | (6u << 22)            // pad_interval: 128 dwords
              | (1u << 25);           // pad_amount: 2 dwords
        g1[1] = ((u32)H2 & 0xFFFFu) << 16;   // tensor_dim0 = 128  (bits 79:48 lo)
        g1[2] = ((u32)H1 & 0xFFFFu) << 16;   // tensor_dim1 = 256  (bits 111:80 lo)
        g1[3] = ((u32)H2) << 16;             // tile_dim0   = 128  (bits 127:112)
        g1[4] = (u32)H1;                     // tile_dim1   = 256  (bits 143:128)
        g1[5] = (u32)H2;                     // tensor_dim0_stride = 128 (bits 207:160 lo)
        g1[6] = 0u;
        g1[7] = 0u;
        const u32x4 gz = {0u, 0u, 0u, 0u};   // groups 2/3: dims 2..4 unused
        asm volatile("tensor_load_to_lds %0, %1, %2, %3"
                     :: "s"(g0), "s"(g1), "s"(gz), "s"(gz)
                     : "memory");
    }

    if (t < 48) sP[t] = points[row0 * 3 + t];
    __syncthreads();

    // ---- layer 1: feat = relu(points @ w_enc + b_enc)   (K=3, VALU) ----
    for (int idx = t; idx < 16 * EE; idx += 256) {
        const int r = idx / EE, c = idx % EE;
        float a = b_enc[c];
        a += sP[r * 3 + 0] * w_enc[0 * EE + c];
        a += sP[r * 3 + 1] * w_enc[1 * EE + c];
        a += sP[r * 3 + 2] * w_enc[2 * EE + c];
        sF[r * FS + c] = fmaxf(a, 0.0f);
    }
    __syncthreads();

    const int lane  = t & 31;
    const int wv    = t >> 5;
    const int half  = lane >> 4;     // 0: K pair (0,1) / rows 0..7 ; 1: K pair (2,3) / rows 8..15
    const int m     = lane & 15;     // A-matrix row
    const int nlane = lane & 15;     // B/D column within tile

    // ---- layer 2: h1 = relu(feat @ w_d1 + b_d1)  (384->256, WMMA fp32, B from L2) ----
#pragma unroll
    for (int ti = 0; ti < 2; ++ti) {
        const int n0 = (wv * 2 + ti) * 16;
        v8f acc = {};
        for (int k0 = 0; k0 < EE; k0 += 4) {
            const int ka = k0 + half * 2;
            const v2f a = *(const v2f*)&sF[m * FS + ka];
            v2f bf;
            bf.x = w_d1[(ka + 0) * H1 + n0 + nlane];
            bf.y = w_d1[(ka + 1) * H1 + n0 + nlane];
            acc = __builtin_amdgcn_wmma_f32_16x16x4_f32(
                false, a, false, bf, (short)0, acc, false, false);
        }
        const int col  = n0 + nlane;
        const float bs = b_d1[col];
#pragma unroll
        for (int r = 0; r < 8; ++r)
            sH1[(r + half * 8) * H1S + col] = fmaxf(acc[r] + bs, 0.0f);
    }
    __syncthreads();

    // TDM data must have landed before layer 3 reads sW2
    if (t < 32) __builtin_amdgcn_s_wait_tensorcnt(0);
    __syncthreads();

    // ---- layer 3: h2 = relu(h1 @ w_d2 + b_d2)  (256->128, WMMA fp32, B from LDS) ----
    {
        const int n0 = wv * 16;
        v8f acc = {};
        for (int k0 = 0; k0 < H1; k0 += 4) {
            const int ka = k0 + half * 2;
            const v2f a = *(const v2f*)&sH1[m * H1S + ka];
            v2f bf;
            bf.x = sW2[(ka + 0) * W2S + n0 + nlane];
            bf.y = sW2[(ka + 1) * W2S + n0 + nlane];
            acc = __builtin_amdgcn_wmma_f32_16x16x4_f32(
                false, a, false, bf, (short)0, acc, false, false);
        }
        const int col  = n0 + nlane;
        const float bs = b_d2[col];
#pragma unroll
        for (int r = 0; r < 8; ++r)
            sH2[(r + half * 8) * H2S + col] = fmaxf(acc[r] + bs, 0.0f);
    }
    __syncthreads();

    // ---- layer 4: rec = h2 @ w_d3 + b_d3  (128->4, VALU) ----
    if (t < 64) {
        const int r = t >> 2, j = t & 3;
        float a = b_d3[j];
        for (int k = 0; k < H2; ++k) a += sH2[r * H2S + k] * w_d3[k * 4 + j];
        sRec[r * 4 + j] = a;
    }
    __syncthreads();

    // ---- loss partials ----
    if (t < 16) {
        const int gi = row0 + t;
        const float w  = wmask[gi];
        const float p0 = sP[t * 3 + 0], p1 = sP[t * 3 + 1], p2 = sP[t * 3 + 2];
        const float r0 = sRec[t * 4 + 0], r1 = sRec[t * 4 + 1];
        const float r2 = sRec[t * 4 + 2], r3 = sRec[t * 4 + 3];

        const float pe = (r0 - p0) * (r0 - p0) + (r1 - p1) * (r1 - p1) + (r2 - p2) * (r2 - p2);

        const float nx = normals[gi * 3 + 0], ny = normals[gi * 3 + 1], nz = normals[gi * 3 + 2];
        const float rn = sqrtf(r0 * r0 + r1 * r1 + r2 * r2);
        const float cs = (r0 * nx + r1 * ny + r2 * nz) / (rn + EPSV);

        const float dc = r3 - curv[gi];

        sRed[0 * 16 + t] = w;
        sRed[1 * 16 + t] = w * pe;
        sRed[2 * 16 + t] = w * cs;
        sRed[3 * 16 + t] = w * dc * dc;
    }
    __syncthreads();
    if (t == 0) {
        float s0 = 0, s1 = 0, s2 = 0, s3 = 0;
        for (int k = 0; k < 16; ++k) {
            s0 += sRed[0 * 16 + k]; s1 += sRed[1 * 16 + k];
            s2 += sRed[2 * 16 + k]; s3 += sRed[3 * 16 + k];
        }
        atomicAdd(&accum[0], s0);
        atomicAdd(&accum[1], s1);
        atomicAdd(&accum[2], s2);
        atomicAdd(&accum[3], s3);
    }
}

// ---------------------------------------------------------------------------
// Kernel 5: combine accumulators into the scalar loss
// ---------------------------------------------------------------------------
__global__ void gcp_final_kernel(const float* __restrict__ accum, float* __restrict__ out) {
    if (threadIdx.x == 0 && blockIdx.x == 0) {
        const float wsum = accum[0] + EPSV;
        const float pos  = accum[1] / (wsum * 3.0f);
        const float nl   = 1.0f - accum[2] / wsum;
        const float cl   = accum[3] / wsum;
        out[0] = pos + 0.5f * nl + 0.2f * cl;
    }
}

// ---------------------------------------------------------------------------
extern "C" void kernel_launch(void* const* d_in, const int* in_sizes, int n_in,
                              void* d_out, int out_size, void* d_ws, size_t ws_size,
                              hipStream_t stream) {
    const float* points    = (const float*)d_in[0];
    const float* mask_rand = (const float*)d_in[1];
    const float* w_enc     = (const float*)d_in[2];
    const float* b_enc     = (const float*)d_in[3];
    const float* w_d1      = (const float*)d_in[4];
    const float* b_d1      = (const float*)d_in[5];
    const float* w_d2      = (const float*)d_in[6];
    const float* b_d2      = (const float*)d_in[7];
    const float* w_d3      = (const float*)d_in[8];
    const float* b_d3      = (const float*)d_in[9];
    float* out = (float*)d_out;

    // workspace layout (~832 KB)
    char* ws = (char*)d_ws;
    float* accum   = (float*)ws;                       // 4 floats
    int*   knn_idx = (int*)  (ws + 256);               // BN*8 ints = 512 KB
    float* wmask   = (float*)(ws + 256 + (size_t)BN * KNN * 4);
    float* normals = (float*)(ws + 256 + (size_t)BN * KNN * 4 + (size_t)BN * 4);
    float* curvat  = (float*)(ws + 256 + (size_t)BN * KNN * 4 + (size_t)BN * 4 + (size_t)BN * 3 * 4);

    gcp_init_kernel<<<1, 32, 0, stream>>>(accum);
    gcp_knn_kernel<<<BN / KT, KT, 0, stream>>>(points, mask_rand, knn_idx, wmask);
    gcp_normals_kernel<<<BN / 128, 128, 0, stream>>>(points, knn_idx, normals);
    gcp_curv_kernel<<<BN / 128, 128, 0, stream>>>(normals, knn_idx, curvat);
    gcp_mlp_loss_kernel<<<BN / 16, 256, 0, stream>>>(
        points, w_enc, b_enc, w_d1, b_d1, w_d2, b_d2, w_d3, b_d3,
        wmask, normals, curvat, accum);
    gcp_final_kernel<<<1, 32, 0, stream>>>(accum, out);
}